// JoCoR_31387620999224
// MI455X (gfx1250) — compile-verified
//
#include <hip/hip_runtime.h>
#include <hip/hip_bf16.h>
#include <stdint.h>

#define NBLK  2500
#define TPB   256
#define WAVES (TPB / 32)
#define PIPE  4

// ---------------------------------------------------------------------------
// CDNA5 async global->LDS copy (gfx1250). One wave-wide instruction: each lane
// supplies a 32-bit LDS byte offset (VDST) and a 64-bit global VA (VADDR pair).
// Tracked by ASYNCcnt; drained with s_wait_asynccnt.
// ---------------------------------------------------------------------------
__device__ __forceinline__ void async_ld_b128(uint32_t lds_off, const void* gaddr) {
    asm volatile("global_load_async_to_lds_b128 %0, %1, off"
                 :
                 : "v"(lds_off), "v"(gaddr)
                 : "memory");
}

#define WAIT_ASYNC(N) asm volatile("s_wait_asynccnt " #N ::: "memory")

// ---------------------------------------------------------------------------
// Per-element JoCoR loss:
//   bce = softplus(x1) - x1*y + softplus(x2) - x2*y
//   p_i = clip(sigmoid(x_i), 1e-4, 1-1e-4)
//   kl  = p2*(log p2 - p1) + (1-p2)*(log(1-p2) - (1-p1))
//       + p1*(log p1 - p2) + (1-p1)*(log(1-p1) - (1-p2))
//   total = 0.1*bce + 0.9*kl
// Uses: softplus(x) = max(x,0) + log1p(exp(-|x|)),
//       log(sigmoid(x)) = x - softplus(x),  log(1-sigmoid(x)) = -softplus(x).
// Clamp region (|x| > 9.2103) handled with constant selects, no extra trans.
// ---------------------------------------------------------------------------
__device__ __forceinline__ float jocor_elem(float x1, float x2, float y) {
    const float EPSV     = 1e-4f;
    const float ONEME    = 1.0f - 1e-4f;
    const float LOG_EPS  = -9.210340371976182f;      // log(1e-4)
    const float LOG_1ME  = -1.0000500033334732e-4f;  // log(1 - 1e-4)

    float e1  = __expf(-fabsf(x1));
    float e2  = __expf(-fabsf(x2));
    float l1  = __logf(1.0f + e1);
    float l2  = __logf(1.0f + e2);
    float sp1 = fmaxf(x1, 0.0f) + l1;                // softplus(x1)
    float sp2 = fmaxf(x2, 0.0f) + l2;                // softplus(x2)

    float bce = (sp1 - x1 * y) + (sp2 - x2 * y);

    float d1 = 1.0f / (1.0f + e1);                   // sigmoid(|x1|)
    float d2 = 1.0f / (1.0f + e2);
    float s1 = (x1 >= 0.0f) ? d1 : (1.0f - d1);      // sigmoid(x1)
    float s2 = (x2 >= 0.0f) ? d2 : (1.0f - d2);

    float lp1 = x1 - sp1, lq1 = -sp1;
    float lp2 = x2 - sp2, lq2 = -sp2;
    float p1 = s1, p2 = s2;

    if (s1 < EPSV)       { p1 = EPSV;  lp1 = LOG_EPS; lq1 = LOG_1ME; }
    else if (s1 > ONEME) { p1 = ONEME; lp1 = LOG_1ME; lq1 = LOG_EPS; }
    if (s2 < EPSV)       { p2 = EPSV;  lp2 = LOG_EPS; lq2 = LOG_1ME; }
    else if (s2 > ONEME) { p2 = ONEME; lp2 = LOG_1ME; lq2 = LOG_EPS; }

    float q1 = 1.0f - p1, q2 = 1.0f - p2;
    float kl = p2 * (lp2 - p1) + q2 * (lq2 - q1)
             + p1 * (lp1 - p2) + q1 * (lq1 - q2);

    return 0.1f * bce + 0.9f * kl;
}

// ---------------------------------------------------------------------------
// Kernel 1: streaming fused loss + block-level partial sums.
// Per-wave async double(quad)-buffered pipeline, no barriers in hot loop.
// ---------------------------------------------------------------------------
__global__ void __launch_bounds__(TPB)
jocor_partial_kernel(const float* __restrict__ x1, const float* __restrict__ x2,
                     const float* __restrict__ yv, float* __restrict__ partial, int n) {
    __shared__ float4 stage[PIPE][3][WAVES][32];   // 48 KB
    __shared__ float  wsum[WAVES];

    const int tid  = threadIdx.x;
    const int lane = tid & 31;
    const int wv   = tid >> 5;
    const int gtid = blockIdx.x * TPB + tid;
    const int TOT  = NBLK * TPB;                   // float4 slots per stage
    const int n4   = n >> 2;
    const int nstages = n4 / TOT;

    const float4* g1 = (const float4*)x1;
    const float4* g2 = (const float4*)x2;
    const float4* gy = (const float4*)yv;

    float acc = 0.0f;

    auto issue = [&](int s) {
        int idx = s * TOT + gtid;
        int buf = s & (PIPE - 1);
        async_ld_b128((uint32_t)(uintptr_t)&stage[buf][0][wv][lane], g1 + idx);
        async_ld_b128((uint32_t)(uintptr_t)&stage[buf][1][wv][lane], g2 + idx);
        async_ld_b128((uint32_t)(uintptr_t)&stage[buf][2][wv][lane], gy + idx);
    };

    int issued = 0;
    for (int p = 0; p < PIPE - 1 && p < nstages; ++p) { issue(p); ++issued; }

    for (int t = 0; t < nstages; ++t) {
        if (issued < nstages) { issue(issued); ++issued; }
        int pend = issued - t - 1;        // stages allowed to stay in flight
        if      (pend >= 3) { WAIT_ASYNC(9); }
        else if (pend == 2) { WAIT_ASYNC(6); }
        else if (pend == 1) { WAIT_ASYNC(3); }
        else                { WAIT_ASYNC(0); }

        int buf = t & (PIPE - 1);
        float4 a = stage[buf][0][wv][lane];
        float4 b = stage[buf][1][wv][lane];
        float4 c = stage[buf][2][wv][lane];
        acc += jocor_elem(a.x, b.x, c.x) + jocor_elem(a.y, b.y, c.y)
             + jocor_elem(a.z, b.z, c.z) + jocor_elem(a.w, b.w, c.w);
    }

    // Remainder float4s (direct global loads; empty for 4096x5000).
    for (int i = nstages * TOT + gtid; i < n4; i += TOT) {
        float4 a = g1[i], b = g2[i], c = gy[i];
        acc += jocor_elem(a.x, b.x, c.x) + jocor_elem(a.y, b.y, c.y)
             + jocor_elem(a.z, b.z, c.z) + jocor_elem(a.w, b.w, c.w);
    }
    // Scalar tail (n not divisible by 4; empty here).
    for (int i = (n4 << 2) + gtid; i < n; i += TOT) {
        acc += jocor_elem(x1[i], x2[i], yv[i]);
    }

    // wave32 reduction
    for (int o = 16; o > 0; o >>= 1) acc += __shfl_down(acc, o, 32);
    if (lane == 0) wsum[wv] = acc;
    __syncthreads();
    if (wv == 0) {
        float v = (lane < WAVES) ? wsum[lane] : 0.0f;
        for (int o = 4; o > 0; o >>= 1) v += __shfl_down(v, o, 32);
        if (lane == 0) partial[blockIdx.x] = v;
    }
}

// ---------------------------------------------------------------------------
// Kernel 2: deterministic finalize — sum block partials, write mean.
// ---------------------------------------------------------------------------
__global__ void __launch_bounds__(TPB)
jocor_final_kernel(const float* __restrict__ partial, int nblk, float inv_n,
                   float* __restrict__ out) {
    __shared__ float wsum[WAVES];
    const int lane = threadIdx.x & 31;
    const int wv   = threadIdx.x >> 5;

    float acc = 0.0f;
    for (int i = threadIdx.x; i < nblk; i += TPB) acc += partial[i];
    for (int o = 16; o > 0; o >>= 1) acc += __shfl_down(acc, o, 32);
    if (lane == 0) wsum[wv] = acc;
    __syncthreads();
    if (wv == 0) {
        float v = (lane < WAVES) ? wsum[lane] : 0.0f;
        for (int o = 4; o > 0; o >>= 1) v += __shfl_down(v, o, 32);
        if (lane == 0) out[0] = v * inv_n;
    }
}

extern "C" void kernel_launch(void* const* d_in, const int* in_sizes, int n_in,
                              void* d_out, int out_size, void* d_ws, size_t ws_size,
                              hipStream_t stream) {
    (void)n_in; (void)out_size; (void)ws_size;
    const float* x1 = (const float*)d_in[0];
    const float* x2 = (const float*)d_in[1];
    const float* yv = (const float*)d_in[2];
    const int n = in_sizes[0];          // 4096 * 5000

    float* partial = (float*)d_ws;      // NBLK floats of scratch

    jocor_partial_kernel<<<NBLK, TPB, 0, stream>>>(x1, x2, yv, partial, n);
    jocor_final_kernel<<<1, TPB, 0, stream>>>(partial, NBLK, 1.0f / (float)n,
                                              (float*)d_out);
}